// GraphConvolution_8435315769432
// MI455X (gfx1250) — compile-verified
//
#include <hip/hip_runtime.h>
#include <hip/hip_bf16.h>

typedef __attribute__((ext_vector_type(2))) float v2f;
typedef __attribute__((ext_vector_type(8))) float v8f;

#define GC_N_NODES 100000
#define GC_FEAT 64

// ---------------------------------------------------------------------------
// Kernel 1: zero the support accumulator (d_ws is poisoned, must clear).
// ---------------------------------------------------------------------------
__global__ void gc_zero_kernel(float4* __restrict__ p, int n4) {
    int i = blockIdx.x * blockDim.x + threadIdx.x;
    if (i < n4) p[i] = make_float4(0.f, 0.f, 0.f, 0.f);
}

// ---------------------------------------------------------------------------
// Kernel 2: SpMM scatter.  32 lanes per edge, each lane owns 2 features.
// x[col] rows are L2-resident (25.6 MB << 192 MB L2), atomics land in L2.
// unsafeAtomicAdd -> native GLOBAL_ATOMIC_ADD_F32 (no CAS loop).
// ---------------------------------------------------------------------------
__global__ void gc_spmm_scatter(const float*  __restrict__ x,
                                const int*    __restrict__ erow,
                                const int*    __restrict__ ecol,
                                const float*  __restrict__ eval,
                                float*        __restrict__ support,
                                int E) {
    int tid  = blockIdx.x * blockDim.x + threadIdx.x;
    int e    = tid >> 5;
    int lane = tid & 31;
    if (e >= E) return;

    int   r = erow[e];
    int   c = ecol[e];
    float v = eval[e];

    const float2* xs = (const float2*)(x + (size_t)c * GC_FEAT);
    float2 m = xs[lane];                       // features 2*lane, 2*lane+1
    float* dst = support + (size_t)r * GC_FEAT + lane * 2;
    unsafeAtomicAdd(dst,     v * m.x);
    unsafeAtomicAdd(dst + 1, v * m.y);
}

// ---------------------------------------------------------------------------
// Kernel 3: out = support @ weight + bias, then row-wise L2 normalize.
// One wave32 per 16-row block; 4x v8f accumulators cover all 64 output cols.
// V_WMMA_F32_16X16X4_F32, K unrolled 16x (K=64).
//
// Fragment layout (ISA 7.12.2, f32 16x16x4):
//   A: lane L holds A[m = L%16][k = kb, kb+1], kb = 2*(L/16)  -> float2 load
//   B: lane L holds B[k = kb, kb+1][n = L%16]
//   C/D: reg v, lane L  ->  row (v + 8*(L/16)), col (L%16)
// ---------------------------------------------------------------------------
__global__ void gc_gemm_norm(const float* __restrict__ support,
                             const float* __restrict__ weight,
                             const float* __restrict__ bias,
                             float*       __restrict__ out,
                             int nTiles) {
    int wavesPerBlock = blockDim.x >> 5;
    int rowTile = blockIdx.x * wavesPerBlock + ((int)threadIdx.x >> 5);
    if (rowTile >= nTiles) return;            // uniform per wave: EXEC stays all-1s

    int lane = threadIdx.x & 31;
    int m    = lane & 15;                     // row (A) / col (B,C)
    int hi   = lane >> 4;                     // half-wave select
    int kb   = hi * 2;                        // k sub-offset within K=4 step

    int rowBase = rowTile * 16;
    v8f acc0 = {}, acc1 = {}, acc2 = {}, acc3 = {};

    const float* arow = support + (size_t)(rowBase + m) * GC_FEAT + kb;

#pragma unroll
    for (int ki = 0; ki < 16; ++ki) {
        int k = ki * 4 + kb;
        float2 av = *(const float2*)(arow + ki * 4);   // A[m][k], A[m][k+1]
        v2f a; a.x = av.x; a.y = av.y;

        const float* w0 = weight + (size_t)k * GC_FEAT;        // B row k
        const float* w1 = w0 + GC_FEAT;                        // B row k+1
        v2f b;

        b.x = w0[m];      b.y = w1[m];
        acc0 = __builtin_amdgcn_wmma_f32_16x16x4_f32(false, a, false, b,
                                                     (short)0, acc0, false, false);
        b.x = w0[m + 16]; b.y = w1[m + 16];
        acc1 = __builtin_amdgcn_wmma_f32_16x16x4_f32(false, a, false, b,
                                                     (short)0, acc1, false, false);
        b.x = w0[m + 32]; b.y = w1[m + 32];
        acc2 = __builtin_amdgcn_wmma_f32_16x16x4_f32(false, a, false, b,
                                                     (short)0, acc2, false, false);
        b.x = w0[m + 48]; b.y = w1[m + 48];
        acc3 = __builtin_amdgcn_wmma_f32_16x16x4_f32(false, a, false, b,
                                                     (short)0, acc3, false, false);
    }

    // bias add (col = t*16 + m)
    float b0 = bias[m], b1 = bias[m + 16], b2 = bias[m + 32], b3 = bias[m + 48];
#pragma unroll
    for (int v = 0; v < 8; ++v) {
        acc0[v] += b0; acc1[v] += b1; acc2[v] += b2; acc3[v] += b3;
    }

    // Row-wise L2 norm: reg v + half hi -> row (v + 8*hi); its 64 cols live in
    // the 16 lanes of this half across acc0..acc3.  Butterfly-reduce each half.
    float inv[8];
#pragma unroll
    for (int v = 0; v < 8; ++v) {
        float p = acc0[v] * acc0[v] + acc1[v] * acc1[v]
                + acc2[v] * acc2[v] + acc3[v] * acc3[v];
        p += __shfl_xor(p, 1, 32);
        p += __shfl_xor(p, 2, 32);
        p += __shfl_xor(p, 4, 32);
        p += __shfl_xor(p, 8, 32);            // all 16 lanes of half hold row sumsq
        inv[v] = 1.0f / __builtin_sqrtf(p);   // matches reference out/norm (no eps)
    }

#pragma unroll
    for (int v = 0; v < 8; ++v) {
        float* orow = out + (size_t)(rowBase + v + 8 * hi) * GC_FEAT;
        float s = inv[v];
        orow[m]      = acc0[v] * s;
        orow[m + 16] = acc1[v] * s;
        orow[m + 32] = acc2[v] * s;
        orow[m + 48] = acc3[v] * s;
    }
}

// ---------------------------------------------------------------------------
extern "C" void kernel_launch(void* const* d_in, const int* in_sizes, int n_in,
                              void* d_out, int out_size, void* d_ws, size_t ws_size,
                              hipStream_t stream) {
    const float* x      = (const float*)d_in[0];
    const int*   erow   = (const int*)  d_in[1];
    const int*   ecol   = (const int*)  d_in[2];
    const float* eval   = (const float*)d_in[3];
    const float* weight = (const float*)d_in[4];
    const float* bias   = (const float*)d_in[5];
    float*       out    = (float*)d_out;
    float*       support = (float*)d_ws;           // N*64 f32 = 25.6 MB scratch

    int E  = in_sizes[1];                          // 1,600,000
    int nSup = GC_N_NODES * GC_FEAT;               // 6.4M floats

    // 1) clear accumulator
    {
        int n4 = nSup / 4;
        gc_zero_kernel<<<(n4 + 255) / 256, 256, 0, stream>>>((float4*)support, n4);
    }
    // 2) scatter-add SpMM (32 lanes/edge)
    {
        long long threads = (long long)E * 32;
        int blocks = (int)((threads + 255) / 256);
        gc_spmm_scatter<<<blocks, 256, 0, stream>>>(x, erow, ecol, eval, support, E);
    }
    // 3) WMMA GEMM + bias + L2 normalize (1 wave per 16 rows, 8 waves/block)
    {
        int nTiles = GC_N_NODES / 16;              // 6250
        int wavesPerBlock = 8;
        int blocks = (nTiles + wavesPerBlock - 1) / wavesPerBlock;
        gc_gemm_norm<<<blocks, wavesPerBlock * 32, 0, stream>>>(support, weight, bias, out, nTiles);
    }
}